// WeightedAttentionPooling_33371895890591
// MI455X (gfx1250) — compile-verified
//
#include <hip/hip_runtime.h>

typedef float v2f __attribute__((ext_vector_type(2)));
typedef float v8f __attribute__((ext_vector_type(8)));

// ---- monotonic float <-> uint mapping so we can use integer atomicMax -----
__device__ __forceinline__ unsigned int fenc(float x) {
    unsigned int u = __float_as_uint(x);
    return (u & 0x80000000u) ? ~u : (u | 0x80000000u);
}
__device__ __forceinline__ float fdec(unsigned int u) {
    unsigned int b = (u & 0x80000000u) ? (u & 0x7FFFFFFFu) : ~u;
    return __uint_as_float(b);
}
#define GMAX_INIT 0x007FFFFFu /* fenc(-inf) */

// ---------------------------------------------------------------------------
// Pass 0: init segment scratch
// ---------------------------------------------------------------------------
__global__ void wap_init(unsigned int* __restrict__ gmax_enc,
                         float* __restrict__ denom,
                         float* __restrict__ numer, int nseg) {
    int s = blockIdx.x * blockDim.x + threadIdx.x;
    if (s < nseg) {
        gmax_enc[s] = GMAX_INIT;
        denom[s] = 0.f;
        numer[s] = 0.f;
    }
}

// ---------------------------------------------------------------------------
// Pass A: per 16-row tile of x, compute gate/msg with fp32 WMMA:
//   D(16x16) = A(16x4) x B(4x16) + C accumulated over K=128.
//   A = [Wg-chunk ; Wm-chunk ; zeros...],  B = x-tile^T chunk.
//   => c[0] lanes 0..15 = gate(row lane), c[1] lanes 0..15 = msg(row lane).
//
// Operand layout (ISA 7.12.2, 32-bit 16x4 / 4x16):
//   A: lane L<16 holds A[L][0..1],  lane L>=16 holds A[L-16][2..3]
//   B: lane n<16 holds B[0..1][n],  lane n>=16 holds B[2..3][n-16]
// so per K-step k0 a lane needs exactly two contiguous float2s of its row:
//   cols {k0+2*half, +1} (frag0) and {k0+2*half+4, +1} (frag1) -> direct b64
//   loads, no cross-lane shuffles. A fragments are tile-invariant and are
//   preloaded once per wave into registers.
// ---------------------------------------------------------------------------
__global__ __launch_bounds__(256)
void wap_pass_a(const float* __restrict__ x,
                const float* __restrict__ Wg,
                const float* __restrict__ Wm,
                const float* __restrict__ bgp,
                const float* __restrict__ bmp,
                const long long* __restrict__ index,
                float* __restrict__ gate_out,
                float* __restrict__ msg_out,
                unsigned int* __restrict__ gmax_enc,
                int num_tiles) {
    const int lane = threadIdx.x & 31;
    const int half = lane >> 4;   // 0: K pairs {0,1}; 1: K pairs {2,3}
    const int n16  = lane & 15;

    // Tile-invariant A fragments: row 0 of A = Wg, row 1 = Wm, rows 2..15 = 0.
    v2f a0f[16], a1f[16];
    const float* wsrc = (n16 == 1) ? Wm : Wg;
    #pragma unroll
    for (int j = 0; j < 16; ++j) {
        const int kk = (j << 3) + (half << 1);
        v2f z; z.x = 0.f; z.y = 0.f;
        a0f[j] = z; a1f[j] = z;
        if (n16 < 2) {
            a0f[j] = *(const v2f*)(wsrc + kk);
            a1f[j] = *(const v2f*)(wsrc + kk + 4);
        }
    }
    const float bg = bgp[0];
    const float bm = bmp[0];

    const int wave   = blockIdx.x * (blockDim.x >> 5) + (threadIdx.x >> 5);
    const int nwaves = gridDim.x * (blockDim.x >> 5);

    for (int t = wave; t < num_tiles; t += nwaves) {
        const int rowbase = t << 4;
        const float* xrow = x + (size_t)(rowbase + n16) * 128 + (half << 1);
        v8f c = {0.f, 0.f, 0.f, 0.f, 0.f, 0.f, 0.f, 0.f};
        #pragma unroll
        for (int j = 0; j < 16; ++j) {
            const int k0 = j << 3;  // K = k0..k0+7 this iteration (2 WMMAs)
            v2f b0 = *(const v2f*)(xrow + k0);       // B frag, K k0..k0+3
            v2f b1 = *(const v2f*)(xrow + k0 + 4);   // B frag, K k0+4..k0+7
            c = __builtin_amdgcn_wmma_f32_16x16x4_f32(false, a0f[j], false, b0,
                                                      (short)0, c, false, false);
            c = __builtin_amdgcn_wmma_f32_16x16x4_f32(false, a1f[j], false, b1,
                                                      (short)0, c, false, false);
        }
        if (lane < 16) {
            const int row = rowbase + lane;
            const float gate = c[0] + bg;  // D[0][lane]
            const float msg  = c[1] + bm;  // D[1][lane]
            gate_out[row] = gate;
            msg_out[row]  = msg;
            const int seg = (int)index[row];
            atomicMax(&gmax_enc[seg], fenc(gate));
        }
    }
}

// scalar tail for rows beyond the last full 16-row tile (N%16, normally 0)
__global__ void wap_pass_a_tail(const float* __restrict__ x,
                                const float* __restrict__ Wg,
                                const float* __restrict__ Wm,
                                const float* __restrict__ bgp,
                                const float* __restrict__ bmp,
                                const long long* __restrict__ index,
                                float* __restrict__ gate_out,
                                float* __restrict__ msg_out,
                                unsigned int* __restrict__ gmax_enc,
                                int start, int n) {
    int i = start + blockIdx.x * blockDim.x + threadIdx.x;
    if (i >= n) return;
    const float* r = x + (size_t)i * 128;
    float g = 0.f, m = 0.f;
    for (int k = 0; k < 128; ++k) { g += r[k] * Wg[k]; m += r[k] * Wm[k]; }
    g += bgp[0]; m += bmp[0];
    gate_out[i] = g; msg_out[i] = m;
    atomicMax(&gmax_enc[(int)index[i]], fenc(g));
}

// ---------------------------------------------------------------------------
// Pass B: e = w^p * exp(gate - gmax); keyed segmented wave scan (index sorted)
// so only segment tails touch global atomics.
// ---------------------------------------------------------------------------
__global__ __launch_bounds__(256)
void wap_pass_b(const float* __restrict__ gate_in,
                const float* __restrict__ msg_in,
                const float* __restrict__ weights,
                const float* __restrict__ pp,
                const long long* __restrict__ index,
                const unsigned int* __restrict__ gmax_enc,
                float* __restrict__ denom,
                float* __restrict__ numer, int n) {
    const float pv = pp[0];
    const int lane = threadIdx.x & 31;
    const int tid = blockIdx.x * blockDim.x + threadIdx.x;
    const int T = gridDim.x * blockDim.x;
    for (int i = tid; __any(i < n); i += T) {
        const bool act = (i < n);
        int seg = -1;
        float e = 0.f, em = 0.f;
        if (act) {
            seg = (int)index[i];
            const float gm = fdec(gmax_enc[seg]);
            e = powf(weights[i], pv) * expf(gate_in[i] - gm);
            em = e * msg_in[i];
        }
        // segmented inclusive scan over wave, keys are sorted within the wave
        #pragma unroll
        for (int off = 1; off < 32; off <<= 1) {
            const float ue  = __shfl_up(e, off, 32);
            const float uem = __shfl_up(em, off, 32);
            const int   us  = __shfl_up(seg, off, 32);
            if (lane >= off && us == seg) { e += ue; em += uem; }
        }
        const int ns = __shfl_down(seg, 1, 32);
        if (act && (lane == 31 || ns != seg)) {
            atomicAdd(&denom[seg], e);
            atomicAdd(&numer[seg], em);
        }
    }
}

// ---------------------------------------------------------------------------
// Pass C: finalize
// ---------------------------------------------------------------------------
__global__ void wap_pass_c(const float* __restrict__ denom,
                           const float* __restrict__ numer,
                           float* __restrict__ out, int nseg) {
    int s = blockIdx.x * blockDim.x + threadIdx.x;
    if (s < nseg) out[s] = numer[s] / (denom[s] + 1e-10f);
}

// ---------------------------------------------------------------------------
extern "C" void kernel_launch(void* const* d_in, const int* in_sizes, int n_in,
                              void* d_out, int out_size, void* d_ws, size_t ws_size,
                              hipStream_t stream) {
    const float*     x   = (const float*)d_in[0];
    const float*     w   = (const float*)d_in[1];
    const float*     Wg  = (const float*)d_in[2];
    const float*     bg  = (const float*)d_in[3];
    const float*     Wm  = (const float*)d_in[4];
    const float*     bm  = (const float*)d_in[5];
    const float*     p   = (const float*)d_in[6];
    const long long* idx = (const long long*)d_in[7];

    const int n    = in_sizes[1];  // N rows
    const int nseg = out_size;     // number of segments

    // workspace: gate[N] msg[N] gmax_enc[S] denom[S] numer[S]  (~17.2 MB)
    float*        gate  = (float*)d_ws;
    float*        msg   = gate + n;
    unsigned int* gmax  = (unsigned int*)(msg + n);
    float*        denom = (float*)(gmax + nseg);
    float*        numer = denom + nseg;
    float*        out   = (float*)d_out;

    const dim3 blk(256);
    wap_init<<<(nseg + 255) / 256, blk, 0, stream>>>(gmax, denom, numer, nseg);

    const int tiles = n >> 4;
    if (tiles > 0) {
        int gridA = (tiles + 7) / 8;       // 8 waves per block
        if (gridA > 2048) gridA = 2048;
        wap_pass_a<<<gridA, blk, 0, stream>>>(x, Wg, Wm, bg, bm, idx,
                                              gate, msg, gmax, tiles);
    }
    const int rem_start = tiles << 4;
    if (rem_start < n) {
        wap_pass_a_tail<<<(n - rem_start + 63) / 64, 64, 0, stream>>>(
            x, Wg, Wm, bg, bm, idx, gate, msg, gmax, rem_start, n);
    }

    int gridB = (n + 255) / 256;
    if (gridB > 2048) gridB = 2048;
    wap_pass_b<<<gridB, blk, 0, stream>>>(gate, msg, w, p, idx, gmax,
                                          denom, numer, n);
    wap_pass_c<<<(nseg + 255) / 256, blk, 0, stream>>>(denom, numer, out, nseg);
}